// Transformer_25013889531990
// MI455X (gfx1250) — compile-verified
//
#include <hip/hip_runtime.h>
#include <math.h>

#define CDIM 512          // model dim
#define CN   512          // sequence length
#define CB   4            // batch
#define CH   8            // heads
#define CDH  64           // dim per head
#define CMROWS (CB * CN)             // 2048 rows
#define CMN ((size_t)CMROWS * CDIM)  // 1048576 elems per buffer
#define CEPS 1e-5f
#define CATTSCALE 0.125f  // 64^-0.5

typedef __attribute__((ext_vector_type(2))) float v2f;
typedef __attribute__((ext_vector_type(8))) float v8f;

// fp32 WMMA: D(16x16,f32) = A(16x4,f32) * B(4x16,f32) + C
__device__ __forceinline__ v8f wmma4(v2f a, v2f b, v8f c) {
  return __builtin_amdgcn_wmma_f32_16x16x4_f32(
      /*neg_a=*/false, a, /*neg_b=*/false, b,
      /*c_mod=*/(short)0, c, /*reuse_a=*/false, /*reuse_b=*/false);
}

// ---------------------------------------------------------------------------
// Complex LayerNorm over last dim (512). One wave per (b,n) row.
// ---------------------------------------------------------------------------
__global__ __launch_bounds__(128)
void cln_kernel(const float* __restrict__ xr, const float* __restrict__ xi,
                const float* __restrict__ gr, const float* __restrict__ br,
                const float* __restrict__ gi, const float* __restrict__ bi,
                float* __restrict__ yr, float* __restrict__ yi)
{
  const int lane = threadIdx.x & 31;
  const int row  = (blockIdx.x << 2) + (threadIdx.x >> 5);   // 2048 rows total
  const float* pr = xr + (size_t)row * CDIM;
  const float* pi = xi + (size_t)row * CDIM;
  float vr[16], vi[16], sr = 0.f, si = 0.f;
#pragma unroll
  for (int j = 0; j < 16; ++j) {
    vr[j] = pr[lane + 32 * j]; vi[j] = pi[lane + 32 * j];
    sr += vr[j]; si += vi[j];
  }
#pragma unroll
  for (int m = 16; m >= 1; m >>= 1) {
    sr += __shfl_xor(sr, m, 32); si += __shfl_xor(si, m, 32);
  }
  const float mr = sr * (1.f / CDIM), mi = si * (1.f / CDIM);
  float qr = 0.f, qi = 0.f;
#pragma unroll
  for (int j = 0; j < 16; ++j) {
    const float dr = vr[j] - mr, di = vi[j] - mi;
    qr += dr * dr; qi += di * di;
  }
#pragma unroll
  for (int m = 16; m >= 1; m >>= 1) {
    qr += __shfl_xor(qr, m, 32); qi += __shfl_xor(qi, m, 32);
  }
  const float ivr = rsqrtf(qr * (1.f / CDIM) + CEPS);
  const float ivi = rsqrtf(qi * (1.f / CDIM) + CEPS);
  float* outr = yr + (size_t)row * CDIM;
  float* outi = yi + (size_t)row * CDIM;
#pragma unroll
  for (int j = 0; j < 16; ++j) {
    const int c = lane + 32 * j;
    outr[c] = (vr[j] - mr) * ivr * gr[c] + br[c];
    outi[c] = (vi[j] - mi) * ivi * gi[c] + bi[c];
  }
}

// ---------------------------------------------------------------------------
// Complex GEMM: Y[m,e] = sum_d A[m,d] * W[e,d]   (A: [2048 x 512], W: [512 x 512])
// One wave computes a 32x32 complex output tile (2x2 sub-tiles of 16x16):
//   - A fragments reused across 2 column tiles, B across 2 row tiles
//     -> 8 b64 loads feed 16 WMMAs per k-step (0.5 loads/WMMA)
//   - complex product via negated imag-A chain: 2 accumulator chains/sub-tile
// mode 0: plain store
// mode 1: Y += bias[col] + Res          (out-projection + residual)
// mode 2: Y = CReLU(Res + ss*Y - ss*lm (real only))    (ISTA feed-forward)
// ---------------------------------------------------------------------------
__global__ __launch_bounds__(256)
void cgemm_kernel(const float* __restrict__ Ar, const float* __restrict__ Ai,
                  const float* __restrict__ Wr, const float* __restrict__ Wi,
                  float* __restrict__ Yr, float* __restrict__ Yi,
                  const float* __restrict__ biasr, const float* __restrict__ biasi,
                  const float* __restrict__ Rr, const float* __restrict__ Ri,
                  const float* __restrict__ ssp, const float* __restrict__ lmp,
                  int layer, int mode)
{
  const int lane = threadIdx.x & 31;
  const int tile = blockIdx.x * 8 + (threadIdx.x >> 5);  // 1024 tiles, 8 waves/block
  const int m0 = (tile >> 4) << 5;                       // row-tile * 32  (64 tiles)
  const int n0 = (tile & 15) << 5;                       // col-tile * 32  (16 tiles)
  const int l16  = lane & 15;
  const int koff = (lane >> 4) << 1;                     // A/B lane K pair: 0 or 2

  const float* a0r = Ar + (size_t)(m0 + l16) * CDIM + koff;
  const float* a0i = Ai + (size_t)(m0 + l16) * CDIM + koff;
  const float* a1r = a0r + (size_t)16 * CDIM;
  const float* a1i = a0i + (size_t)16 * CDIM;
  const float* b0r = Wr + (size_t)(n0 + l16) * CDIM + koff;
  const float* b0i = Wi + (size_t)(n0 + l16) * CDIM + koff;
  const float* b1r = b0r + (size_t)16 * CDIM;
  const float* b1i = b0i + (size_t)16 * CDIM;

  // acc_r/acc_i for sub-tiles (sm,sn): index = sm*2+sn
  v8f acr0 = {}, aci0 = {}, acr1 = {}, aci1 = {};
  v8f acr2 = {}, aci2 = {}, acr3 = {}, aci3 = {};
#pragma unroll 2
  for (int k = 0; k < CDIM; k += 4) {
    const v2f A0r = *(const v2f*)(a0r + k);
    const v2f A0i = *(const v2f*)(a0i + k);
    const v2f A1r = *(const v2f*)(a1r + k);
    const v2f A1i = *(const v2f*)(a1i + k);
    const v2f B0r = *(const v2f*)(b0r + k);
    const v2f B0i = *(const v2f*)(b0i + k);
    const v2f B1r = *(const v2f*)(b1r + k);
    const v2f B1i = *(const v2f*)(b1i + k);
    const v2f nA0i = -A0i;
    const v2f nA1i = -A1i;
    // (m0,n0)
    acr0 = wmma4(A0r, B0r, wmma4(nA0i, B0i, acr0));
    aci0 = wmma4(A0r, B0i, wmma4(A0i,  B0r, aci0));
    // (m0,n1)
    acr1 = wmma4(A0r, B1r, wmma4(nA0i, B1i, acr1));
    aci1 = wmma4(A0r, B1i, wmma4(A0i,  B1r, aci1));
    // (m1,n0)
    acr2 = wmma4(A1r, B0r, wmma4(nA1i, B0i, acr2));
    aci2 = wmma4(A1r, B0i, wmma4(A1i,  B0r, aci2));
    // (m1,n1)
    acr3 = wmma4(A1r, B1r, wmma4(nA1i, B1i, acr3));
    aci3 = wmma4(A1r, B1i, wmma4(A1i,  B1r, aci3));
  }

  float ss = 0.f, sl = 0.f;
  if (mode == 2) {
    ss = log1pf(expf(ssp[layer]));                 // softplus(step_size)
    sl = ss * log1pf(expf(lmp[layer]));            // ss * softplus(lambda)
  }
  const v8f* ACR[4] = { &acr0, &acr1, &acr2, &acr3 };
  const v8f* ACI[4] = { &aci0, &aci1, &aci2, &aci3 };
#pragma unroll
  for (int st = 0; st < 4; ++st) {
    const int rowbase = m0 + ((st >> 1) << 4) + ((lane >> 4) << 3);
    const int col     = n0 + ((st & 1) << 4) + l16;
    const v8f ar = *ACR[st];
    const v8f ai = *ACI[st];
#pragma unroll
    for (int r = 0; r < 8; ++r) {
      const size_t idx = (size_t)(rowbase + r) * CDIM + col;
      float yr = ar[r];
      float yi = ai[r];
      if (mode == 1) {
        yr += biasr[col] + Rr[idx];
        yi += biasi[col] + Ri[idx];
      } else if (mode == 2) {
        yr = Rr[idx] + ss * yr - sl;  yr = yr > 0.f ? yr : 0.f;  // complex - real
        yi = Ri[idx] + ss * yi;       yi = yi > 0.f ? yi : 0.f;  // real part only
      }
      Yr[idx] = yr;
      Yi[idx] = yi;
    }
  }
}

// ---------------------------------------------------------------------------
// Fused complex attention for one (b, h, 16-query tile). One wave per block.
// Pass 1: dots = (Q Conj(K)^T)*scale via WMMA -> LDS strip [16 x 512] complex,
//         track row-max of |dots|.
// Pass 2: magnitude-softmax numerators with phase preserved, row sums.
// Pass 3: O = attn * V via WMMA, normalized by row sum in the epilogue.
// ---------------------------------------------------------------------------
__global__ __launch_bounds__(32)
void cattn_kernel(const float* __restrict__ wr, const float* __restrict__ wi,
                  float* __restrict__ outr, float* __restrict__ outi)
{
  __shared__ float ldsDr[16 * CN];   // 32 KB
  __shared__ float ldsDi[16 * CN];   // 32 KB
  const int lane = threadIdx.x;
  const int nt = blockIdx.x & 31;          // query tile
  const int h  = (blockIdx.x >> 5) & 7;    // head
  const int b  = blockIdx.x >> 8;          // batch
  const int n0 = nt << 4;
  const int l16  = lane & 15;
  const int half = lane >> 4;
  const int koff = half << 1;

  const float* baser = wr + (size_t)b * CN * CDIM + h * CDH;
  const float* basei = wi + (size_t)b * CN * CDIM + h * CDH;
  const float* qrp = baser + (size_t)(n0 + l16) * CDIM + koff;
  const float* qip = basei + (size_t)(n0 + l16) * CDIM + koff;

  float rmax[8], rsum[8];
#pragma unroll
  for (int r = 0; r < 8; ++r) { rmax[r] = 0.f; rsum[r] = 0.f; }  // |z| >= 0

  // ---- Pass 1: scores strip ----
  for (int kt = 0; kt < CN / 16; ++kt) {
    const float* krp = baser + (size_t)((kt << 4) + l16) * CDIM + koff;
    const float* kip = basei + (size_t)((kt << 4) + l16) * CDIM + koff;
    v8f a_dr = {}, a_di = {};
#pragma unroll
    for (int k = 0; k < CDH; k += 4) {
      const v2f q_r = *(const v2f*)(qrp + k);
      const v2f q_i = *(const v2f*)(qip + k);
      const v2f k_r = *(const v2f*)(krp + k);
      const v2f k_i = *(const v2f*)(kip + k);
      const v2f nq_r = -q_r;
      // q*conj(k): re = qr*kr + qi*ki,  im = qi*kr - qr*ki
      a_dr = wmma4(q_r, k_r, wmma4(q_i,  k_i, a_dr));
      a_di = wmma4(q_i, k_r, wmma4(nq_r, k_i, a_di));
    }
#pragma unroll
    for (int r = 0; r < 8; ++r) {
      const float dr = a_dr[r] * CATTSCALE;
      const float di = a_di[r] * CATTSCALE;
      const float mag = sqrtf(dr * dr + di * di);
      const int row = r + (half << 3);
      const int col = (kt << 4) + l16;
      ldsDr[row * CN + col] = dr;
      ldsDi[row * CN + col] = di;
      rmax[r] = fmaxf(rmax[r], mag);
    }
  }
#pragma unroll
  for (int r = 0; r < 8; ++r)
#pragma unroll
    for (int m = 1; m < 16; m <<= 1)
      rmax[r] = fmaxf(rmax[r], __shfl_xor(rmax[r], m, 32));
  __syncthreads();

  // ---- Pass 2: softmax on magnitude, phase preserved ----
  for (int kt = 0; kt < CN / 16; ++kt) {
#pragma unroll
    for (int r = 0; r < 8; ++r) {
      const int row = r + (half << 3);
      const int col = (kt << 4) + l16;
      const float dr = ldsDr[row * CN + col];
      const float di = ldsDi[row * CN + col];
      const float mag = sqrtf(dr * dr + di * di);
      const float e = __expf(mag - rmax[r]);
      const float cr = mag > 0.f ? dr / mag : 1.f;   // cos(angle(0)) = 1
      const float ci = mag > 0.f ? di / mag : 0.f;
      rsum[r] += e;
      ldsDr[row * CN + col] = e * cr;
      ldsDi[row * CN + col] = e * ci;
    }
  }
#pragma unroll
  for (int r = 0; r < 8; ++r)
#pragma unroll
    for (int m = 1; m < 16; m <<= 1)
      rsum[r] += __shfl_xor(rsum[r], m, 32);
  __syncthreads();

  // ---- Pass 3: O = attn * V, per 16-wide d chunk ----
  for (int ch = 0; ch < 4; ++ch) {
    v8f o_r = {}, o_i = {};
    for (int kt = 0; kt < CN / 16; ++kt) {
#pragma unroll
      for (int s = 0; s < 4; ++s) {
        const int kk = (kt << 4) + (s << 2) + koff;  // key pair base (even)
        const v2f Aa_r = *(const v2f*)&ldsDr[l16 * CN + kk];
        const v2f Aa_i = *(const v2f*)&ldsDi[l16 * CN + kk];
        const v2f nAa_i = -Aa_i;
        const float* p_r = baser + (size_t)kk * CDIM + (ch << 4) + l16;
        const float* p_i = basei + (size_t)kk * CDIM + (ch << 4) + l16;
        const v2f Bb_r = { p_r[0], p_r[CDIM] };
        const v2f Bb_i = { p_i[0], p_i[CDIM] };
        // attn*v: re = ar*vr - ai*vi, im = ar*vi + ai*vr
        o_r = wmma4(Aa_r, Bb_r, wmma4(nAa_i, Bb_i, o_r));
        o_i = wmma4(Aa_r, Bb_i, wmma4(Aa_i,  Bb_r, o_i));
      }
    }
    const int rowbase = n0 + (half << 3);
    const int col = h * CDH + (ch << 4) + l16;
#pragma unroll
    for (int r = 0; r < 8; ++r) {
      const float inv = 1.f / rsum[r];
      const size_t idx = ((size_t)b * CN + rowbase + r) * CDIM + col;
      outr[idx] = o_r[r] * inv;
      outi[idx] = o_i[r] * inv;
    }
  }
}

// ---------------------------------------------------------------------------
extern "C" void kernel_launch(void* const* d_in, const int* in_sizes, int n_in,
                              void* d_out, int out_size, void* d_ws, size_t ws_size,
                              hipStream_t stream)
{
  (void)in_sizes; (void)n_in; (void)out_size; (void)ws_size;

  const float* x_real  = (const float*)d_in[0];
  const float* x_imag  = (const float*)d_in[1];
  const float* ln1_g_r = (const float*)d_in[2];
  const float* ln1_b_r = (const float*)d_in[3];
  const float* ln1_g_i = (const float*)d_in[4];
  const float* ln1_b_i = (const float*)d_in[5];
  const float* ln2_g_r = (const float*)d_in[6];
  const float* ln2_b_r = (const float*)d_in[7];
  const float* ln2_g_i = (const float*)d_in[8];
  const float* ln2_b_i = (const float*)d_in[9];
  const float* qkv_w_r = (const float*)d_in[10];
  const float* qkv_w_i = (const float*)d_in[11];
  const float* out_w_r = (const float*)d_in[12];
  const float* out_w_i = (const float*)d_in[13];
  const float* out_b_r = (const float*)d_in[14];
  const float* out_b_i = (const float*)d_in[15];
  const float* ff_w_r  = (const float*)d_in[16];
  const float* ff_w_i  = (const float*)d_in[17];
  const float* step_sz = (const float*)d_in[18];
  const float* lambd   = (const float*)d_in[19];

  float* ws  = (float*)d_ws;
  float* xr  = ws + 0 * CMN;   // current activation
  float* xi  = ws + 1 * CMN;
  float* xnr = ws + 2 * CMN;   // LayerNorm output
  float* xni = ws + 3 * CMN;
  float* wqr = ws + 4 * CMN;   // QKV projection output
  float* wqi = ws + 5 * CMN;
  float* aor = ws + 6 * CMN;   // attention output
  float* aoi = ws + 7 * CMN;

  hipMemcpyAsync(xr, x_real, CMN * sizeof(float), hipMemcpyDeviceToDevice, stream);
  hipMemcpyAsync(xi, x_imag, CMN * sizeof(float), hipMemcpyDeviceToDevice, stream);

  const dim3 gLN(512),   bLN(128);   // 2048 rows, 4 waves/block
  const dim3 gGEMM(128), bGEMM(256); // 1024 32x32 tiles, 8 waves/block
  const dim3 gATT(1024), bATT(32);   // B*H*(N/16) wave-tiles

  const size_t DD = (size_t)CDIM * CDIM;
  for (int l = 0; l < 4; ++l) {
    // PreNorm + shared-QKV attention + residual
    cln_kernel<<<gLN, bLN, 0, stream>>>(xr, xi,
        ln1_g_r + l * CDIM, ln1_b_r + l * CDIM,
        ln1_g_i + l * CDIM, ln1_b_i + l * CDIM, xnr, xni);
    cgemm_kernel<<<gGEMM, bGEMM, 0, stream>>>(xnr, xni,
        qkv_w_r + l * DD, qkv_w_i + l * DD, wqr, wqi,
        nullptr, nullptr, nullptr, nullptr, nullptr, nullptr, l, 0);
    cattn_kernel<<<gATT, bATT, 0, stream>>>(wqr, wqi, aor, aoi);
    cgemm_kernel<<<gGEMM, bGEMM, 0, stream>>>(aor, aoi,
        out_w_r + l * DD, out_w_i + l * DD, xr, xi,
        out_b_r + l * CDIM, out_b_i + l * CDIM, xr, xi, nullptr, nullptr, l, 1);
    // PreNorm + ISTA feed-forward (CReLU), no outer residual
    cln_kernel<<<gLN, bLN, 0, stream>>>(xr, xi,
        ln2_g_r + l * CDIM, ln2_b_r + l * CDIM,
        ln2_g_i + l * CDIM, ln2_b_i + l * CDIM, xnr, xni);
    cgemm_kernel<<<gGEMM, bGEMM, 0, stream>>>(xnr, xni,
        ff_w_r + l * DD, ff_w_i + l * DD, xr, xi,
        nullptr, nullptr, xnr, xni, step_sz, lambd, l, 2);
  }

  float* out = (float*)d_out;              // [2, B, N, DIM] fp32
  hipMemcpyAsync(out,       xr, CMN * sizeof(float), hipMemcpyDeviceToDevice, stream);
  hipMemcpyAsync(out + CMN, xi, CMN * sizeof(float), hipMemcpyDeviceToDevice, stream);
}